// SFNet_27376121544789
// MI455X (gfx1250) — compile-verified
//
#include <hip/hip_runtime.h>
#include <stdint.h>

#define B   8
#define H   64
#define W   64
#define HW  4096
#define NCOL (B * HW)          // 32768 (b x target-pixel) columns
#define INV2SIG2 0.02f         // 1 / (2 * 5^2)

typedef __attribute__((ext_vector_type(4))) float v4f;  // native vector: OK for nontemporal builtins

// ---------------- kernel 0: zero the packed argmax buffer ----------------
__global__ void k_init(unsigned long long* __restrict__ packed) {
  int i = blockIdx.x * blockDim.x + threadIdx.x;
  if (i < NCOL) packed[i] = 0ull;
}

// ---------------- kernel 1: partial argmax + packed atomic merge ----------
// grid (NCOL/256, RSPLIT), block 256. One (b,target) column per thread,
// RROWS rows per block-row-chunk. Column stride is 16 KB -> prefetch ahead.
#define RSPLIT 8
#define RROWS  (HW / RSPLIT)   // 512
__global__ void k_argmax(const float* __restrict__ corr,
                         unsigned long long* __restrict__ packed) {
  int colg = blockIdx.x * 256 + threadIdx.x;        // 0..32767
  int b    = colg >> 12;
  int col  = colg & (HW - 1);
  int r0   = blockIdx.y * RROWS;
  const float* p = corr + ((size_t)b * HW) * HW + (size_t)r0 * HW + col;

  float m  = -__builtin_inff();
  int   mi = 0;
#pragma unroll 4
  for (int r = 0; r < RROWS; ++r) {
    __builtin_prefetch(p + (size_t)(r + 16) * HW, 0, 0);  // global_prefetch_b8
    float v = p[(size_t)r * HW];
    if (v > m) { m = v; mi = r0 + r; }                    // strict > : first max wins
  }
  // order-preserving pack; ties across chunks resolve to LOWEST index
  unsigned ord = __float_as_uint(m);
  ord = (ord & 0x80000000u) ? ~ord : (ord | 0x80000000u);
  unsigned long long key = ((unsigned long long)ord << 32) |
                           (unsigned long long)(0xFFFFFFFFu - (unsigned)mi);
  atomicMax(&packed[colg], key);
}

// ---------------- kernel 2: unpack argmax -> (iy<<8)|ix u16 ---------------
__global__ void k_resolve(const unsigned long long* __restrict__ packed,
                          unsigned short* __restrict__ idx16) {
  int i = blockIdx.x * blockDim.x + threadIdx.x;
  if (i >= NCOL) return;
  unsigned idx = 0xFFFFFFFFu - (unsigned)(packed[i] & 0xFFFFFFFFu);
  unsigned iy = (idx >> 6) & 63u, ix = idx & 63u;
  idx16[i] = (unsigned short)((iy << 8) | ix);
}

// ---------------- kernel 3: out = corr * tab[|dy|]*tab[|dx|] --------------
// grid (HW/ROWS_PB, B), block 256. Index table async-staged into LDS.
#define ROWS_PB 32
__global__ void k_apply(const float* __restrict__ corr,
                        const unsigned short* __restrict__ idx16,
                        float* __restrict__ out) {
  __shared__ __align__(16) unsigned short sidx[HW];  // 8 KB per-batch index table
  __shared__ float tab[64];
  const int tid = threadIdx.x;
  const int b   = blockIdx.y;
  // reverse tile order: start pass 2 on the data pass 1 touched last (L2 tail)
  const int row_tile = (gridDim.x - 1 - blockIdx.x);

  if (tid < 64) tab[tid] = expf(-(float)(tid * tid) * INV2SIG2);

  // CDNA5 async global->LDS staging of the 8 KB index table (b128 per lane)
  const unsigned short* gsrc = idx16 + b * HW;
  for (int k = tid; k < HW / 8; k += 256) {
    unsigned lds_off = (unsigned)(uintptr_t)(&sidx[k * 8]);          // LDS byte addr
    unsigned long long ga = (unsigned long long)(uintptr_t)(gsrc + k * 8);
    asm volatile("global_load_async_to_lds_b128 %0, %1, off"
                 :: "v"(lds_off), "v"(ga) : "memory");
  }
  asm volatile("s_wait_asynccnt 0" ::: "memory");
  __syncthreads();

  const size_t base = (size_t)b * HW * HW;
  const int src0 = row_tile * ROWS_PB;
  for (int r = 0; r < ROWS_PB; ++r) {
    const int src = src0 + r;
    const int sy = src >> 6, sx = src & 63;
    const v4f* in4  = (const v4f*)(corr + base + (size_t)src * HW);
    v4f*       out4 = (v4f*)(out  + base + (size_t)src * HW);
#pragma unroll
    for (int j = 0; j < 4; ++j) {
      const int c4 = tid + j * 256;                 // 1024 float4 per row
      __builtin_prefetch(in4 + c4 + (HW / 4), 0, 0);
      v4f v = __builtin_nontemporal_load(&in4[c4]); // last-use read
      const int c = c4 * 4;
      v4f o;
      {
        unsigned pk = sidx[c + 0]; int dy = sy - (int)(pk >> 8), dx = sx - (int)(pk & 255u);
        o.x = v.x * tab[dy < 0 ? -dy : dy] * tab[dx < 0 ? -dx : dx];
      }
      {
        unsigned pk = sidx[c + 1]; int dy = sy - (int)(pk >> 8), dx = sx - (int)(pk & 255u);
        o.y = v.y * tab[dy < 0 ? -dy : dy] * tab[dx < 0 ? -dx : dx];
      }
      {
        unsigned pk = sidx[c + 2]; int dy = sy - (int)(pk >> 8), dx = sx - (int)(pk & 255u);
        o.z = v.z * tab[dy < 0 ? -dy : dy] * tab[dx < 0 ? -dx : dx];
      }
      {
        unsigned pk = sidx[c + 3]; int dy = sy - (int)(pk >> 8), dx = sx - (int)(pk & 255u);
        o.w = v.w * tab[dy < 0 ? -dy : dy] * tab[dx < 0 ? -dx : dx];
      }
      __builtin_nontemporal_store(o, &out4[c4]);    // don't pollute L2 with output
    }
  }
}

// ---------------- host ----------------------------------------------------
extern "C" void kernel_launch(void* const* d_in, const int* in_sizes, int n_in,
                              void* d_out, int out_size, void* d_ws, size_t ws_size,
                              hipStream_t stream) {
  (void)in_sizes; (void)n_in; (void)out_size; (void)ws_size;
  const float* corr = (const float*)d_in[0];
  float*       outp = (float*)d_out;

  unsigned long long* packed = (unsigned long long*)d_ws;                   // 256 KB
  unsigned short*     idx16  = (unsigned short*)((char*)d_ws + NCOL * 8);   // +64 KB

  k_init   <<<NCOL / 256, 256, 0, stream>>>(packed);
  k_argmax <<<dim3(NCOL / 256, RSPLIT), 256, 0, stream>>>(corr, packed);
  k_resolve<<<NCOL / 256, 256, 0, stream>>>(packed, idx16);
  k_apply  <<<dim3(HW / ROWS_PB, B), 256, 0, stream>>>(corr, idx16, outp);
}